// GraphAttentionLayer_57191784513927
// MI455X (gfx1250) — compile-verified
//
#include <hip/hip_runtime.h>
#include <math.h>

// GAT layer, MI455X (gfx1250), wave32. f32 WMMA 16x16x4 for both GEMMs.
#define Bn 4
#define Nn 512
#define Fo 64
#define Ki 128
#define ALPHAF 0.2f

typedef __attribute__((ext_vector_type(2))) float v2f;
typedef __attribute__((ext_vector_type(8))) float v8f;

__device__ __forceinline__ float lrelu(float v) { return v > 0.f ? v : ALPHAF * v; }

// ---------------- Kernel 1: h = x @ W + b  (M=2048, K=128, N=64) ----------------
// One wave per 16x16 output tile. 512 tiles.
__global__ __launch_bounds__(32) void gat_hgemm(const float* __restrict__ x,
                                                const float* __restrict__ W,
                                                const float* __restrict__ bias,
                                                float* __restrict__ h) {
  const int tile = blockIdx.x;
  const int nt = tile & 3;        // 4 N-tiles
  const int mt = tile >> 2;       // 128 M-tiles
  const int lane = threadIdx.x;   // 0..31
  const int l15 = lane & 15;
  const int kg  = lane >> 4;      // 0 or 1 (K half)
  const int row0 = mt * 16;
  const int n0   = nt * 16;

  v8f acc = {};
  for (int k0 = 0; k0 < Ki; k0 += 4) {
    const int ka = k0 + kg * 2;
    // A fragment: A[m=l15][k=ka], A[m=l15][k=ka+1]
    const float* xr = x + (size_t)(row0 + l15) * Ki + ka;
    v2f Af; Af.x = xr[0]; Af.y = xr[1];
    // B fragment: B[k=ka][n=l15], B[k=ka+1][n=l15]
    const float* wr = W + (size_t)ka * Fo + n0 + l15;
    v2f Bf; Bf.x = wr[0]; Bf.y = wr[Fo];
    acc = __builtin_amdgcn_wmma_f32_16x16x4_f32(false, Af, false, Bf,
                                                (short)0, acc, false, false);
  }
  const float bv = bias[n0 + l15];
  const int rbase = row0 + kg * 8;
#pragma unroll
  for (int i = 0; i < 8; ++i) {
    h[(size_t)(rbase + i) * Fo + n0 + l15] = acc[i] + bv;
  }
}

// ---------------- Kernel 2: p[r] = h[r].a0, q[r] = h[r].a1 ----------------
__global__ void gat_pq(const float* __restrict__ h, const float* __restrict__ a,
                       float* __restrict__ p, float* __restrict__ q) {
  const int r = blockIdx.x * blockDim.x + threadIdx.x;
  if (r >= Bn * Nn) return;
  const float* hr = h + (size_t)r * Fo;
  float sp = 0.f, sq = 0.f;
#pragma unroll 4
  for (int f = 0; f < Fo; ++f) {
    sp += hr[f] * a[f];
    sq += hr[f] * a[Fo + f];
  }
  p[r] = sp;
  q[r] = sq;
}

// ---------------- Kernel 3: masked softmax rows -> prob ----------------
// One wave per (b,n) row; lane handles 16 m-values (m = lane + 32*j).
__global__ __launch_bounds__(128) void gat_softmax(const int* __restrict__ adj,
                                                   const float* __restrict__ p,
                                                   const float* __restrict__ q,
                                                   float* __restrict__ prob) {
  const int row  = blockIdx.x * 4 + (threadIdx.x >> 5); // 0..2047
  const int lane = threadIdx.x & 31;
  const int b = row >> 9;        // /512
  const int n = row & (Nn - 1);  // %512
  const int* adjr = adj + (size_t)b * Nn * Nn + (size_t)n * Nn;
  const float* pb = p + b * Nn;
  const float* qb = q + b * Nn;

  float vlo = 0.f, vhi = 0.f;
  if (n < 256) {
    vlo = lrelu(pb[2 * n]     + qb[2 * n]);
    vhi = lrelu(pb[2 * n + 1] + qb[2 * n + 1]);
  }

  float vals[16];
  float mx = -INFINITY;
#pragma unroll
  for (int j = 0; j < 16; ++j) {
    const int m = lane + 32 * j;
    float e;
    if (n < 256) {
      e = (m < 256) ? vlo : vhi;
    } else {
      const int mm = m & 255;
      e = lrelu(pb[2 * mm] + qb[2 * mm + 1]);
    }
    e = (adjr[m] > 0) ? e : -INFINITY;
    vals[j] = e;
    mx = fmaxf(mx, e);
  }
#pragma unroll
  for (int off = 16; off >= 1; off >>= 1) mx = fmaxf(mx, __shfl_xor(mx, off, 32));

  float sum = 0.f;
#pragma unroll
  for (int j = 0; j < 16; ++j) {
    const float e = vals[j];
    const float ex = (e == -INFINITY) ? 0.f : __expf(e - mx);
    vals[j] = ex;
    sum += ex;
  }
#pragma unroll
  for (int off = 16; off >= 1; off >>= 1) sum += __shfl_xor(sum, off, 32);

  const float inv = 1.f / sum;
  float* pr = prob + (size_t)b * Nn * Nn + (size_t)n * Nn;
#pragma unroll
  for (int j = 0; j < 16; ++j) pr[lane + 32 * j] = vals[j] * inv;
}

// ---------------- Kernel 4: out = ELU(prob @ h)  (per b: 512x512x64) ----------------
// One wave per 16x16 output tile. 4*32*4 = 512 tiles.
__global__ __launch_bounds__(32) void gat_aggr(const float* __restrict__ prob,
                                               const float* __restrict__ h,
                                               float* __restrict__ out) {
  const int tile = blockIdx.x;
  const int nt = tile & 3;
  const int mt = (tile >> 2) & 31;
  const int b  = tile >> 7;
  const int lane = threadIdx.x;
  const int l15 = lane & 15;
  const int kg  = lane >> 4;
  const int row0 = mt * 16;
  const int n0   = nt * 16;

  const float* pb = prob + (size_t)b * Nn * Nn;
  const float* hb = h + (size_t)b * Nn * Fo;

  v8f acc = {};
  for (int k0 = 0; k0 < Nn; k0 += 4) {
    const int ka = k0 + kg * 2;
    const float* ar = pb + (size_t)(row0 + l15) * Nn + ka;
    v2f Af; Af.x = ar[0]; Af.y = ar[1];
    const float* br = hb + (size_t)ka * Fo + n0 + l15;
    v2f Bf; Bf.x = br[0]; Bf.y = br[Fo];
    acc = __builtin_amdgcn_wmma_f32_16x16x4_f32(false, Af, false, Bf,
                                                (short)0, acc, false, false);
  }
  float* ob = out + (size_t)b * Nn * Fo;
  const int rbase = row0 + kg * 8;
#pragma unroll
  for (int i = 0; i < 8; ++i) {
    float v = acc[i];
    v = (v > 0.f) ? v : (__expf(v) - 1.f);   // ELU, alpha=1
    ob[(size_t)(rbase + i) * Fo + n0 + l15] = v;
  }
}

extern "C" void kernel_launch(void* const* d_in, const int* in_sizes, int n_in,
                              void* d_out, int out_size, void* d_ws, size_t ws_size,
                              hipStream_t stream) {
  const float* x    = (const float*)d_in[0];
  const int*   adj  = (const int*)d_in[1];
  const float* W    = (const float*)d_in[2];
  const float* bias = (const float*)d_in[3];
  const float* a    = (const float*)d_in[4];
  float* out = (float*)d_out;

  // Workspace layout (floats): h[2048*64], p[2048], q[2048], prob[4*512*512]
  float* h    = (float*)d_ws;
  float* p    = h + (size_t)Bn * Nn * Fo;
  float* q    = p + Bn * Nn;
  float* prob = q + Bn * Nn;

  gat_hgemm<<<512, 32, 0, stream>>>(x, W, bias, h);
  gat_pq<<<(Bn * Nn + 255) / 256, 256, 0, stream>>>(h, a, p, q);
  gat_softmax<<<512, 128, 0, stream>>>(adj, p, q, prob);
  gat_aggr<<<512, 32, 0, stream>>>(prob, h, out);
}